// TimesBlock_16801912062039
// MI455X (gfx1250) — compile-verified
//
#include <hip/hip_runtime.h>
#include <hip/hip_bf16.h>

typedef float v2f __attribute__((ext_vector_type(2)));
typedef float v8f __attribute__((ext_vector_type(8)));

// Problem constants
#define NB 4
#define NC 48
#define NOC 48
#define NT 24
#define NS 4096           // H*W
#define BN_EPS 1e-5f

// Workspace layout (floats)
#define PERI_OFF 0                      // 4608
#define BETA_OFF 4608                   // 48
#define QMAT_OFF 4672                   // 48*48*9 = 20736
#define KFRG_OFF 25408                  // 3*5*12*32*2 = 11520
// total 36928 floats = 147712 bytes

// Main kernel tiling: 8 waves x 64 columns = 512-column segments
#define SEG 512
#define XI  (SEG + 4)                   // 516 staged columns (2 halo each side)
#define CP  52                          // padded channel stride in LDS ([s][c] layout)
#define A_FLOATS (3*5*12*32*2)          // 11520
#define X_OFF A_FLOATS
#define SMEM_FLOATS (A_FLOATS + XI*CP)  // 11520 + 26832 = 38352
#define SMEM_BYTES (SMEM_FLOATS * 4)    // 153408 B  (< 320KB/WGP, 2 WGs resident)

// ---------------------------------------------------------------- kernel 1
// peri[b,c,t] = mean over h*w of x
__global__ void k_pool(const float* __restrict__ x, float* __restrict__ peri) {
    int bct = blockIdx.x;                     // (b*48+c)*24+t
    const float* p = x + (size_t)bct * NS;
    float s = 0.f;
    for (int i = threadIdx.x; i < NS; i += 128) s += p[i];
    __shared__ float red[128];
    red[threadIdx.x] = s;
    __syncthreads();
    for (int off = 64; off > 0; off >>= 1) {
        if (threadIdx.x < off) red[threadIdx.x] += red[threadIdx.x + off];
        __syncthreads();
    }
    if (threadIdx.x == 0) peri[bct] = red[0] * (1.f / NS);
}

// ---------------------------------------------------------------- kernel 2
// conv1d(k=3,pad=1)+bias -> BN(batch stats)+ReLU -> mean_t -> alpha -> beta[o]
__global__ void k_small(const float* __restrict__ peri,
                        const float* __restrict__ pw,   // (48,48,3)
                        const float* __restrict__ pb,   // (48)
                        const float* __restrict__ gam,  // (48)
                        const float* __restrict__ bet,  // (48)
                        const float* __restrict__ aw,   // (48,48,1)
                        float* __restrict__ beta_out) { // (48)
    __shared__ float peS[NB*NC*NT];
    __shared__ float pe2[NB*NC*NT];
    __shared__ float mu[NC], rs[NC];
    __shared__ float pooled[NB*NC];
    __shared__ float alpha[NB*NOC];
    int tid = threadIdx.x;
    for (int i = tid; i < NB*NC*NT; i += 256) peS[i] = peri[i];
    __syncthreads();
    for (int i = tid; i < NB*NC*NT; i += 256) {
        int b = i / (NC*NT); int r = i % (NC*NT); int c = r / NT; int t = r % NT;
        float a = pb[c];
        for (int c2 = 0; c2 < NC; ++c2) {
            const float* w = &pw[(c*NC + c2)*3];
            int base = (b*NC + c2)*NT;
            if (t > 0)      a += w[0] * peS[base + t - 1];
                            a += w[1] * peS[base + t];
            if (t < NT - 1) a += w[2] * peS[base + t + 1];
        }
        pe2[i] = a;
    }
    __syncthreads();
    if (tid < NC) {
        float m = 0.f;
        for (int b = 0; b < NB; ++b)
            for (int t = 0; t < NT; ++t) m += pe2[(b*NC + tid)*NT + t];
        m *= (1.f / (NB*NT));
        float v = 0.f;
        for (int b = 0; b < NB; ++b)
            for (int t = 0; t < NT; ++t) {
                float d = pe2[(b*NC + tid)*NT + t] - m; v += d * d;
            }
        v *= (1.f / (NB*NT));
        mu[tid] = m; rs[tid] = rsqrtf(v + BN_EPS);
    }
    __syncthreads();
    if (tid < NB*NC) {
        int b = tid / NC, c = tid % NC;
        float g = gam[c], be = bet[c], m = mu[c], r = rs[c];
        float s = 0.f;
        for (int t = 0; t < NT; ++t) {
            float z = g * (pe2[(b*NC + c)*NT + t] - m) * r + be;
            s += (z > 0.f) ? z : 0.f;
        }
        pooled[tid] = s * (1.f / NT);
    }
    __syncthreads();
    if (tid < NB*NOC) {
        int b = tid / NOC, o = tid % NOC;
        float a = 1.f;
        for (int c = 0; c < NC; ++c) a += aw[o*NC + c] * pooled[b*NC + c];
        alpha[tid] = a;
    }
    __syncthreads();
    if (tid < NOC) {
        float s = 0.f;
        for (int b = 0; b < NB; ++b) { float a = alpha[b*NOC + tid]; s += a * a; }
        beta_out[tid] = s;
    }
}

// ---------------------------------------------------------------- kernel 3
// Q[c'][c][m][k] = sum_o beta[o]*Wi[o,c',m]*Wi[o,c,k]
// K5 written directly in WMMA A-fragment layout [mt][j][kb][lane][v]
__global__ void k_qk(const float* __restrict__ beta,
                     const float* __restrict__ Wi,     // (1,48,48,1,3)
                     float* __restrict__ Q,
                     float* __restrict__ kfrg) {
    int cp = blockIdx.x;
    int c = threadIdx.x;
    if (c >= NC) return;
    float q[3][3] = {{0.f,0.f,0.f},{0.f,0.f,0.f},{0.f,0.f,0.f}};
    for (int o = 0; o < NOC; ++o) {
        float bo = beta[o];
        const float* wa = &Wi[(o*NC + cp)*3];
        const float* wb = &Wi[(o*NC + c)*3];
        float a0 = bo*wa[0], a1 = bo*wa[1], a2 = bo*wa[2];
        for (int k = 0; k < 3; ++k) {
            float w = wb[k];
            q[0][k] += a0*w; q[1][k] += a1*w; q[2][k] += a2*w;
        }
    }
    for (int m = 0; m < 3; ++m)
        for (int k = 0; k < 3; ++k)
            Q[((cp*NC + c)*3 + m)*3 + k] = q[m][k];
    // K5[j] = sum_m q[m][j-2+m], tap offset d = j-2 (x index s+d)
    int mt = cp >> 4, lo = cp & 15;
    int kb = c >> 2, rem = c & 3, hi = rem >> 1, v = rem & 1;
    int lane = hi*16 + lo;
    for (int j = 0; j < 5; ++j) {
        float s = 0.f;
        for (int m = 0; m < 3; ++m) {
            int k = j - 2 + m;
            if (k >= 0 && k < 3) s += q[m][k];
        }
        kfrg[(((mt*5 + j)*12 + kb)*32 + lane)*2 + v] = s;
    }
}

// ---------------------------------------------------------------- kernel 4
// Main WMMA kernel: per (b,t,seg): OUT(48 x 512) = x + sum_j K5_j(48x48) @ X_shift
// Each wave: 4 N-tiles of 16 columns; A fragment reused 4x per load.
__global__ __launch_bounds__(256) void k_main(const float* __restrict__ x,
                                              const float* __restrict__ kfrg,
                                              float* __restrict__ out) {
    extern __shared__ float sm[];
    int b = blockIdx.z, t = blockIdx.y, seg = blockIdx.x;
    int s0 = seg * SEG;
    int tid = threadIdx.x;

    // stage A-fragments (identity copy, layout prebuilt by k_qk)
    for (int i = tid; i < A_FLOATS; i += 256) sm[i] = kfrg[i];
    // stage X transposed: sm[X_OFF + i*CP + c] = x[b,c,t, s0-2+i], zero-padded
    for (int idx = tid; idx < NC*XI; idx += 256) {
        int c = idx / XI, i = idx % XI;
        int s = s0 - 2 + i;
        float v = 0.f;
        if (s >= 0 && s < NS)
            v = x[((size_t)((b*NC + c)*NT + t))*NS + s];
        sm[X_OFF + i*CP + c] = v;
    }
    __syncthreads();

    int lane = tid & 31, wid = tid >> 5;
    int lo = lane & 15, hi = lane >> 4;
    int wcol = wid * 64;                 // this wave's 64-column span (4 N-tiles)

    for (int mt = 0; mt < 3; ++mt) {
        v8f acc0 = {}, acc1 = {}, acc2 = {}, acc3 = {};
        // keep j rolled: bounds live fragments to one tap (12 A + 48 B pairs),
        // stays under 256 VGPRs (no s_set_vgpr_msb churn)
        #pragma unroll 1
        for (int j = 0; j < 5; ++j) {
            const float* abase = &sm[((mt*5 + j)*12)*64 + lane*2];
            const float* xrow  = &sm[X_OFF + (wcol + lo + j)*CP + 2*hi];
            #pragma unroll
            for (int kb = 0; kb < 12; ++kb) {
                v2f A = *(const v2f*)(abase + kb*64);        // ds_load_b64
                const float* xb = xrow + kb*4;
                v2f B0 = *(const v2f*)(xb);
                acc0 = __builtin_amdgcn_wmma_f32_16x16x4_f32(
                        false, A, false, B0, (short)0, acc0, false, false);
                v2f B1 = *(const v2f*)(xb + 16*CP);
                acc1 = __builtin_amdgcn_wmma_f32_16x16x4_f32(
                        false, A, false, B1, (short)0, acc1, false, false);
                v2f B2 = *(const v2f*)(xb + 32*CP);
                acc2 = __builtin_amdgcn_wmma_f32_16x16x4_f32(
                        false, A, false, B2, (short)0, acc2, false, false);
                v2f B3 = *(const v2f*)(xb + 48*CP);
                acc3 = __builtin_amdgcn_wmma_f32_16x16x4_f32(
                        false, A, false, B3, (short)0, acc3, false, false);
            }
        }
        // store with residual (D layout: VGPR v -> M = v + 8*hi, N = lane%16)
        #pragma unroll
        for (int v = 0; v < 8; ++v) {
            int cp = mt*16 + v + 8*hi;
            int sA = s0 + wcol + lo;
            size_t base = ((size_t)((b*NC + cp)*NT + t))*NS;
            float r0 = sm[X_OFF + (wcol + lo +  2)*CP + cp];
            float r1 = sm[X_OFF + (wcol + lo + 18)*CP + cp];
            float r2 = sm[X_OFF + (wcol + lo + 34)*CP + cp];
            float r3 = sm[X_OFF + (wcol + lo + 50)*CP + cp];
            out[base + sA]      = acc0[v] + r0;
            out[base + sA + 16] = acc1[v] + r1;
            out[base + sA + 32] = acc2[v] + r2;
            out[base + sA + 48] = acc3[v] + r3;
        }
    }
}

// ---------------------------------------------------------------- kernel 5
// Exact boundary fixup at s in {0,1,4094,4095}:
// out = x + sum_{m: 0<=s-m<=4093} sum_k sum_c Q[c',c,m,k]*x[c, s-m+k]
__global__ void k_fix(const float* __restrict__ x,
                      const float* __restrict__ Q,
                      float* __restrict__ out) {
    int si = blockIdx.x, t = blockIdx.y, b = blockIdx.z;
    int s = (si < 2) ? si : (NS - 4 + si);          // 0,1,4094,4095
    int cp = threadIdx.x;
    if (cp >= NC) return;
    size_t rowb = ((size_t)((b*NC + cp)*NT + t))*NS;
    float acc = x[rowb + s];
    for (int m = 0; m < 3; ++m) {
        int u = s - m;
        if (u < 0 || u > NS - 3) continue;
        for (int k = 0; k < 3; ++k) {
            int xs = u + k;
            float a = 0.f;
            for (int c = 0; c < NC; ++c)
                a += Q[((cp*NC + c)*3 + m)*3 + k] *
                     x[((size_t)((b*NC + c)*NT + t))*NS + xs];
            acc += a;
        }
    }
    out[rowb + s] = acc;
}

// ---------------------------------------------------------------- launch
extern "C" void kernel_launch(void* const* d_in, const int* in_sizes, int n_in,
                              void* d_out, int out_size, void* d_ws, size_t ws_size,
                              hipStream_t stream) {
    const float* x   = (const float*)d_in[0];
    const float* Wi  = (const float*)d_in[1];
    const float* pw  = (const float*)d_in[2];
    const float* pb  = (const float*)d_in[3];
    const float* gam = (const float*)d_in[4];
    const float* bet = (const float*)d_in[5];
    const float* aw  = (const float*)d_in[6];
    float* out = (float*)d_out;
    float* ws  = (float*)d_ws;

    k_pool<<<NB*NC*NT, 128, 0, stream>>>(x, ws + PERI_OFF);
    k_small<<<1, 256, 0, stream>>>(ws + PERI_OFF, pw, pb, gam, bet, aw,
                                   ws + BETA_OFF);
    k_qk<<<NC, 64, 0, stream>>>(ws + BETA_OFF, Wi, ws + QMAT_OFF, ws + KFRG_OFF);

    (void)hipFuncSetAttribute((const void*)k_main,
                              hipFuncAttributeMaxDynamicSharedMemorySize,
                              SMEM_BYTES);
    k_main<<<dim3(NS/SEG, NT, NB), 256, SMEM_BYTES, stream>>>(x, ws + KFRG_OFF, out);
    k_fix<<<dim3(4, NT, NB), 64, 0, stream>>>(x, ws + QMAT_OFF, out);
}